// MessagePassingLayerV1_86260123173218
// MI455X (gfx1250) — compile-verified
//
#include <hip/hip_runtime.h>

// Problem constants (from reference): B=2, N=50000, D=128, E=500000
#define B_ 2
#define N_ 50000
#define D_ 128
#define E_ 500000
#define M_ (B_ * N_)          // 100000 flattened rows; 100000 = 3125 * 32 exactly
#define LN_EPS 1e-5f

typedef __attribute__((ext_vector_type(2))) float v2f;
typedef __attribute__((ext_vector_type(8))) float v8f;

// ---------------------------------------------------------------------------
// Kernel 1: zero the workspace (agg[B,N,D] + cnt[N]); total divisible by 4.
// ---------------------------------------------------------------------------
__global__ void zero_ws_kernel(float4* __restrict__ ws, int n4) {
    int i = blockIdx.x * blockDim.x + threadIdx.x;
    if (i < n4) ws[i] = make_float4(0.f, 0.f, 0.f, 0.f);
}

// ---------------------------------------------------------------------------
// Kernel 2: edge scatter.  agg[b,dst,:] += h[b,src,:];  cnt[dst] += 1.
// One wave per edge; lane covers 4 channels via float4 load + 4 f32 atomics.
// h and agg both fit in the 192MB L2 -> this is L2-atomic bound, no HBM churn.
// ---------------------------------------------------------------------------
__global__ __launch_bounds__(256) void scatter_kernel(
        const float* __restrict__ h, const int* __restrict__ ei,
        float* __restrict__ agg, float* __restrict__ cnt) {
    int e    = blockIdx.x * (blockDim.x >> 5) + (threadIdx.x >> 5);
    int lane = threadIdx.x & 31;
    if (e >= E_) return;
    int src = ei[e];
    int dst = ei[E_ + e];
    if (lane == 0) atomicAdd(&cnt[dst], 1.0f);
#pragma unroll
    for (int b = 0; b < B_; ++b) {
        const float4 v = *(const float4*)(h + ((size_t)(b * N_ + src)) * D_ + lane * 4);
        float* p = agg + ((size_t)(b * N_ + dst)) * D_ + lane * 4;
        atomicAdd(p + 0, v.x);
        atomicAdd(p + 1, v.y);
        atomicAdd(p + 2, v.z);
        atomicAdd(p + 3, v.w);
    }
}

// ---------------------------------------------------------------------------
// Kernel 3: fused dual-GEMM (f32 WMMA 16x16x4) + residual + ReLU + LayerNorm.
//
// out = LN( h + relu( h@Ws^T + (agg/cnt)@Wm^T + bias ) ) * gamma + beta
//
// K-concatenated formulation: A = [h | agg*rs] (16 x 256 per wave),
// B = [Ws ; Wm] as K x N, so both GEMMs accumulate into ONE chain of 8 v8f
// accumulators and the h_self + h_msg add is free.
//
// Per block: 64 threads = 2 waves, each wave owns 16 rows (32 rows/block,
// 3125 blocks, no tail).  Weights staged to LDS in 64-wide K chunks with a
// 68-float pitch (bank stride 4 -> conflict-free fragment reads).
//
// Fragment layouts per CDNA5 ISA (CDNA5 ISA):
//   A 16x4 f32 : lane m=L%16; VGPR0/1 = K = 4kb + 2*(L>=16) + {0,1}
//   B 4x16 f32 : assumed mirror of A (n=L%16 across lanes, same K split)
//   C 16x16    : VGPR v -> row v + 8*(L>=16), col L%16 (+16 per n-tile)
// ---------------------------------------------------------------------------
__global__ __launch_bounds__(64) void fused_gemm_ln_kernel(
        const float* __restrict__ h, const float* __restrict__ agg,
        const float* __restrict__ cnt,
        const float* __restrict__ Wself, const float* __restrict__ Wmsg,
        const float* __restrict__ bias, const float* __restrict__ gamma,
        const float* __restrict__ beta, float* __restrict__ out) {
    __shared__ float Wl[128 * 68];   // 34,816 B of LDS (320KB available/WGP)

    const int tid  = threadIdx.x;
    const int lane = tid & 31;
    const int wave = tid >> 5;
    const int half = (lane >> 4) & 1;   // 0: lanes 0-15, 1: lanes 16-31
    const int l16  = lane & 15;

    const int rowbase = blockIdx.x * 32 + wave * 16;   // 16 rows per wave
    const int gm      = rowbase + l16;                 // A row this lane feeds
    const int node    = gm % N_;
    const float rs    = 1.0f / fmaxf(cnt[node], 1.0f); // mean clamp(count,1)

    v8f acc[8] = {};

    const float* Arow_h = h   + (size_t)gm * D_;
    const float* Arow_a = agg + (size_t)gm * D_;

    for (int chunk = 0; chunk < 4; ++chunk) {
        // Stage B[kk][n] = W[n][k] chunk: Wl[n*68 + kl] = Wsrc[n*128 + koff + kl]
        const float* Wsrc = (chunk < 2) ? Wself : Wmsg;
        const int koff    = (chunk & 1) * 64;
        __syncthreads();
        for (int i = tid; i < 128 * 16; i += 64) {      // uniform trip count
            int n = i >> 4, f4 = i & 15;
            float4 w = *(const float4*)(Wsrc + n * 128 + koff + f4 * 4);
            *(float4*)(&Wl[n * 68 + f4 * 4]) = w;
        }
        __syncthreads();

        const bool isMsg = (chunk >= 2);
#pragma unroll
        for (int kb = 0; kb < 16; ++kb) {
            const int kl0 = kb * 4 + half * 2;
            v2f a;
            if (!isMsg) {
                a = *(const v2f*)(Arow_h + koff + kl0);
            } else {
                v2f t2 = *(const v2f*)(Arow_a + koff + kl0);
                a.x = t2.x * rs;
                a.y = t2.y * rs;
            }
#pragma unroll
            for (int t = 0; t < 8; ++t) {
                const int n = l16 + 16 * t;
                v2f b = *(const v2f*)(&Wl[n * 68 + kl0]);
                acc[t] = __builtin_amdgcn_wmma_f32_16x16x4_f32(
                    false, a, false, b, (short)0, acc[t], false, false);
            }
        }
    }

    // ---- Epilogue: residual + ReLU + LayerNorm, all in registers ----------
    float bia[8], gam[8], bet[8];
#pragma unroll
    for (int t = 0; t < 8; ++t) {
        int ct = l16 + 16 * t;
        bia[t] = bias[ct]; gam[t] = gamma[ct]; bet[t] = beta[ct];
    }

    const int rbase = rowbase + half * 8;
    float s1[8], s2[8];
#pragma unroll
    for (int v = 0; v < 8; ++v) { s1[v] = 0.f; s2[v] = 0.f; }

#pragma unroll
    for (int t = 0; t < 8; ++t) {
        int ct = l16 + 16 * t;
#pragma unroll
        for (int v = 0; v < 8; ++v) {
            int gr   = rbase + v;
            float hv = h[(size_t)gr * D_ + ct];
            float x  = hv + fmaxf(acc[t][v] + bia[t], 0.0f);
            acc[t][v] = x;
            s1[v] += x;
            s2[v] += x * x;
        }
    }
    // Row sums live across the 16-lane half-group -> xor-shuffle reduce.
#pragma unroll
    for (int v = 0; v < 8; ++v) {
#pragma unroll
        for (int m = 1; m < 16; m <<= 1) {
            s1[v] += __shfl_xor(s1[v], m, 32);
            s2[v] += __shfl_xor(s2[v], m, 32);
        }
    }
#pragma unroll
    for (int v = 0; v < 8; ++v) {
        float mu   = s1[v] * (1.0f / 128.0f);
        float var  = s2[v] * (1.0f / 128.0f) - mu * mu;
        float rstd = rsqrtf(var + LN_EPS);
        int gr = rbase + v;
#pragma unroll
        for (int t = 0; t < 8; ++t) {
            float y = (acc[t][v] - mu) * rstd * gam[t] + bet[t];
            out[(size_t)gr * D_ + (l16 + 16 * t)] = y;
        }
    }
}

// ---------------------------------------------------------------------------
extern "C" void kernel_launch(void* const* d_in, const int* in_sizes, int n_in,
                              void* d_out, int out_size, void* d_ws, size_t ws_size,
                              hipStream_t stream) {
    const float* h     = (const float*)d_in[0];
    const int*   ei    = (const int*)d_in[1];   // [2,E]: src row then dst row
    const float* Wself = (const float*)d_in[2];
    const float* Wmsg  = (const float*)d_in[3];
    const float* bias  = (const float*)d_in[4];
    const float* gamma = (const float*)d_in[5];
    const float* beta  = (const float*)d_in[6];
    float* out = (float*)d_out;

    float* agg = (float*)d_ws;                       // [B,N,D] f32 = 51.2 MB
    float* cnt = agg + (size_t)B_ * N_ * D_;         // [N] f32

    const int n4 = (B_ * N_ * D_ + N_) / 4;          // divisible by 4
    zero_ws_kernel<<<(n4 + 255) / 256, 256, 0, stream>>>((float4*)d_ws, n4);
    scatter_kernel<<<E_ / 8, 256, 0, stream>>>(h, ei, agg, cnt);
    fused_gemm_ln_kernel<<<M_ / 32, 64, 0, stream>>>(h, agg, cnt, Wself, Wmsg,
                                                     bias, gamma, beta, out);
}